// IntergraphInteract_33560874451730
// MI455X (gfx1250) — compile-verified
//
#include <hip/hip_runtime.h>
#include <hip/hip_bf16.h>

#define NQ_   10000
#define NT_   20000
#define D_    256
#define E_    500000
#define C_    1000
#define EPS_  1e-10f

typedef __attribute__((ext_vector_type(2))) float v2f;
typedef __attribute__((ext_vector_type(8))) float v8f;

// ---------------------------------------------------------------------------
// 0) init workspace: Acc = 0, norm = 0, winner = -1
// ---------------------------------------------------------------------------
__global__ void init_ws_kernel(float* __restrict__ acc,
                               float* __restrict__ norm,
                               int*   __restrict__ winner) {
    int i = blockIdx.x * blockDim.x + threadIdx.x;
    if (i < NT_ * D_) acc[i] = 0.0f;
    if (i < NT_) { norm[i] = 0.0f; winner[i] = -1; }
}

// ---------------------------------------------------------------------------
// 1) Y = Xq @ W   (NQ x D) = (NQ x D)(D x D), f32 WMMA 16x16x4
//    One wave per 16x16 output tile; K loop over D in steps of 4.
//    A layout (16x4 f32): lanes 0-15 -> M=lane, v0=K0 v1=K1 ;
//                         lanes 16-31 -> M=lane-16, v0=K2 v1=K3.
//    B layout (4x16 f32): lanes 0-15 -> N=lane, v0=K0 v1=K1 ;
//                         lanes 16-31 -> N=lane-16, v0=K2 v1=K3.
//    C/D (16x16 f32): VGPR r, lanes 0-15 -> M=r, lanes 16-31 -> M=8+r; N=lane%16.
// ---------------------------------------------------------------------------
__global__ __launch_bounds__(128) void gemm_y_kernel(const float* __restrict__ Xq,
                                                     const float* __restrict__ W,
                                                     float* __restrict__ Y) {
    const int wave = blockIdx.x * (blockDim.x >> 5) + (threadIdx.x >> 5);
    const int lane = threadIdx.x & 31;
    const int nTilesN = D_ / 16;                  // 16
    const int tm = wave / nTilesN;                // 0..624
    const int tn = wave % nTilesN;                // 0..15
    if (tm >= NQ_ / 16) return;

    const int m0 = tm * 16, n0 = tn * 16;
    const int lhalf = lane >> 4;                  // 0 or 1
    const int l16   = lane & 15;

    const float* arow = Xq + (m0 + l16) * D_ + lhalf * 2;       // +k0 -> K, +1 -> K+1
    const float* bcol = W  + (lhalf * 2) * D_ + (n0 + l16);     // +k0*D -> K row

    v8f c = {0.f, 0.f, 0.f, 0.f, 0.f, 0.f, 0.f, 0.f};

    #pragma unroll 4
    for (int k0 = 0; k0 < D_; k0 += 4) {
        v2f a, b;
        a.x = arow[k0];
        a.y = arow[k0 + 1];
        b.x = bcol[k0 * D_];
        b.y = bcol[k0 * D_ + D_];
        c = __builtin_amdgcn_wmma_f32_16x16x4_f32(
                /*neg_a=*/false, a, /*neg_b=*/false, b,
                /*c_mod=*/(short)0, c, /*reuse_a=*/false, /*reuse_b=*/false);
    }

    float* yout = Y + (m0 + lhalf * 8) * D_ + (n0 + l16);
    #pragma unroll
    for (int r = 0; r < 8; ++r) yout[r * D_] = c[r];
}

// ---------------------------------------------------------------------------
// 2) per-edge: score = dot(Y[u], Xt[v]) + b ; w = exp(score); beta = sigmoid;
//    atomically accumulate msg into Acc[v], w into norm[v].
//    One wave (32 lanes) per edge; each lane owns 8 contiguous floats of D=256.
// ---------------------------------------------------------------------------
__global__ __launch_bounds__(256) void edge_kernel(const float* __restrict__ Xq,
                                                   const float* __restrict__ Xt,
                                                   const float* __restrict__ Y,
                                                   const float* __restrict__ bptr,
                                                   const int*   __restrict__ u_idx,
                                                   const int*   __restrict__ v_idx,
                                                   float* __restrict__ acc,
                                                   float* __restrict__ norm) {
    const int e    = blockIdx.x * (blockDim.x >> 5) + (threadIdx.x >> 5);
    const int lane = threadIdx.x & 31;
    if (e >= E_) return;

    const int u = u_idx[e];
    const int v = v_idx[e];

    const float* yp = Y  + u * D_ + lane * 8;
    const float* tp = Xt + v * D_ + lane * 8;
    const float4 y0 = *(const float4*)(yp);
    const float4 y1 = *(const float4*)(yp + 4);
    const float4 t0 = *(const float4*)(tp);
    const float4 t1 = *(const float4*)(tp + 4);

    float dot = y0.x * t0.x + y0.y * t0.y + y0.z * t0.z + y0.w * t0.w
              + y1.x * t1.x + y1.y * t1.y + y1.z * t1.z + y1.w * t1.w;
    #pragma unroll
    for (int off = 16; off > 0; off >>= 1)
        dot += __shfl_xor(dot, off, 32);

    const float score = dot + bptr[0];
    const float w     = expf(score);
    const float beta  = 1.0f / (1.0f + expf(-score));
    const float cq    = w * (1.0f - beta);
    const float ct    = w * beta;

    const float* qp = Xq + u * D_ + lane * 8;
    const float4 q0 = *(const float4*)(qp);
    const float4 q1 = *(const float4*)(qp + 4);

    float* ap = acc + v * D_ + lane * 8;
    unsafeAtomicAdd(ap + 0, cq * q0.x + ct * t0.x);
    unsafeAtomicAdd(ap + 1, cq * q0.y + ct * t0.y);
    unsafeAtomicAdd(ap + 2, cq * q0.z + ct * t0.z);
    unsafeAtomicAdd(ap + 3, cq * q0.w + ct * t0.w);
    unsafeAtomicAdd(ap + 4, cq * q1.x + ct * t1.x);
    unsafeAtomicAdd(ap + 5, cq * q1.y + ct * t1.y);
    unsafeAtomicAdd(ap + 6, cq * q1.z + ct * t1.z);
    unsafeAtomicAdd(ap + 7, cq * q1.w + ct * t1.w);

    if (lane == 0) unsafeAtomicAdd(norm + v, w);
}

// ---------------------------------------------------------------------------
// 3) out = Acc / (norm + eps)
// ---------------------------------------------------------------------------
__global__ void finalize_kernel(const float* __restrict__ acc,
                                const float* __restrict__ norm,
                                float* __restrict__ out) {
    int i = blockIdx.x * blockDim.x + threadIdx.x;
    if (i < NT_ * D_) {
        int row = i >> 8;   // / D_
        out[i] = acc[i] / (norm[row] + EPS_);
    }
}

// ---------------------------------------------------------------------------
// 4) consensus: last occurrence wins (deterministic) via atomicMax on index
// ---------------------------------------------------------------------------
__global__ void cons_winner_kernel(const int* __restrict__ v_cons,
                                   int* __restrict__ winner) {
    int c = blockIdx.x * blockDim.x + threadIdx.x;
    if (c < C_) atomicMax(&winner[v_cons[c]], c);
}

__global__ void cons_write_kernel(const float* __restrict__ Xq,
                                  const int* __restrict__ u_cons,
                                  const int* __restrict__ v_cons,
                                  const int* __restrict__ winner,
                                  float* __restrict__ out) {
    const int c = blockIdx.x;        // 0..C-1
    const int d = threadIdx.x;       // 0..D-1
    const int v = v_cons[c];
    if (winner[v] == c)
        out[v * D_ + d] = Xq[u_cons[c] * D_ + d];
}

// ---------------------------------------------------------------------------
extern "C" void kernel_launch(void* const* d_in, const int* in_sizes, int n_in,
                              void* d_out, int out_size, void* d_ws, size_t ws_size,
                              hipStream_t stream) {
    const float* Xq    = (const float*)d_in[0];
    const float* Xt    = (const float*)d_in[1];
    const float* W     = (const float*)d_in[2];
    const float* b     = (const float*)d_in[3];
    const int*   u_idx = (const int*)d_in[4];
    const int*   v_idx = (const int*)d_in[5];
    const int*   u_con = (const int*)d_in[6];
    const int*   v_con = (const int*)d_in[7];
    float* out = (float*)d_out;

    // workspace layout
    float* Y      = (float*)d_ws;                 // NQ*D floats
    float* Acc    = Y + NQ_ * D_;                 // NT*D floats
    float* normv  = Acc + NT_ * D_;               // NT floats
    int*   winner = (int*)(normv + NT_);          // NT ints

    // 0) init Acc/norm/winner
    init_ws_kernel<<<(NT_ * D_ + 255) / 256, 256, 0, stream>>>(Acc, normv, winner);

    // 1) Y = Xq @ W  (10000 waves = 625x16 tiles, 4 waves/block)
    {
        const int waves = (NQ_ / 16) * (D_ / 16);
        gemm_y_kernel<<<(waves + 3) / 4, 128, 0, stream>>>(Xq, W, Y);
    }

    // 2) edge scatter (1 wave per edge, 8 waves/block)
    edge_kernel<<<(E_ + 7) / 8, 256, 0, stream>>>(Xq, Xt, Y, b, u_idx, v_idx, Acc, normv);

    // 3) normalize
    finalize_kernel<<<(NT_ * D_ + 255) / 256, 256, 0, stream>>>(Acc, normv, out);

    // 4) consensus overwrite (deterministic last-wins)
    cons_winner_kernel<<<(C_ + 255) / 256, 256, 0, stream>>>(v_con, winner);
    cons_write_kernel<<<C_, D_, 0, stream>>>(Xq, u_con, v_con, winner, out);
}